// GridDeepVPN_33328946217236
// MI455X (gfx1250) — compile-verified
//
#include <hip/hip_runtime.h>

typedef __attribute__((ext_vector_type(2))) float v2f;
typedef __attribute__((ext_vector_type(8))) float v8f;

#define GRID9 9
#define SN 81          // STATE_NUM
#define SN_PAD 84      // K padded to multiple of 4 (zeros)
#define NSC_STRIDE 85  // odd stride -> conflict-free column reads
#define HID 32
#define ROWS_PER_BLOCK 64
#define THREADS 128

__global__ __launch_bounds__(THREADS)
void grid_vpn_fused(const float* __restrict__ obs,
                    const float* __restrict__ ac,
                    const float* __restrict__ W1,
                    const float* __restrict__ W2,
                    const float* __restrict__ W3,
                    const float* __restrict__ b3,
                    float* __restrict__ out,
                    int B)
{
    __shared__ float sW1[SN_PAD][HID];
    __shared__ float sW2[HID][HID];
    __shared__ float sW3[HID];
    __shared__ float sNsc[ROWS_PER_BLOCK][NSC_STRIDE];
    __shared__ float sH[ROWS_PER_BLOCK][HID + 1];
    __shared__ float sIR[ROWS_PER_BLOCK];

    const int tid = threadIdx.x;
    const int b0 = blockIdx.x * ROWS_PER_BLOCK;

    // ---- stage weights into LDS (zero-pad W1 rows 81..83) ----
    for (int k = tid; k < SN_PAD * HID; k += THREADS) {
        int row = k / HID, col = k % HID;
        sW1[row][col] = (row < SN) ? W1[row * HID + col] : 0.0f;
    }
    for (int k = tid; k < HID * HID; k += THREADS)
        sW2[k / HID][k % HID] = W2[k];
    if (tid < HID) sW3[tid] = W3[tid];
    if (tid < ROWS_PER_BLOCK) sIR[tid] = 0.0f;
    // zero the K-pad columns of the nsc tile
    for (int k = tid; k < ROWS_PER_BLOCK * (NSC_STRIDE - SN); k += THREADS) {
        int row = k / (NSC_STRIDE - SN);
        int col = SN + (k % (NSC_STRIDE - SN));
        sNsc[row][col] = 0.0f;
    }
    __syncthreads();

    // ---- gather + segment-sum (stencil over columns), served reward ----
    for (int idx = tid; idx < ROWS_PER_BLOCK * SN; idx += THREADS) {
        int bl = idx / SN, i = idx % SN;
        int b = b0 + bl;
        int r = i / GRID9, c = i % GRID9;
        const float* p = ac + (size_t)b * (SN * SN) + i;   // column i of slab b
        float s = p[(size_t)i * SN];                       // self (first, like ref)
        if (r > 0)         s += p[(size_t)(i - GRID9) * SN];
        if (r < GRID9 - 1) s += p[(size_t)(i + GRID9) * SN];
        if (c > 0)         s += p[(size_t)(i - 1) * SN];
        if (c < GRID9 - 1) s += p[(size_t)(i + 1) * SN];
        sNsc[bl][i] = s;
        out[(size_t)3 * B + (size_t)b * SN + i] = s;       // nsc output
        float demand = obs[(size_t)b * (2 * SN) + SN + i];
        atomicAdd(&sIR[bl], fminf(s, demand));             // ds_add_f32
    }
    __syncthreads();

    // ---- per-wave WMMA MLP over its private 16-row tile ----
    const int wave = tid >> 5;
    const int lane = tid & 31;
    const int m0 = wave * 16;
    const int lm = lane & 15;       // M (A rows) / N (B,C cols)
    const int hi = lane >> 4;       // selects K pair {0,1} vs {2,3}

    // GEMM1: nsc(16x84) @ W1(84x32), f32 WMMA, two N-tiles
    v8f acc0 = {}, acc1 = {};
    for (int kk = 0; kk < SN_PAD / 4; ++kk) {
        int kb = 4 * kk + 2 * hi;
        v2f a;  a.x  = sNsc[m0 + lm][kb];  a.y  = sNsc[m0 + lm][kb + 1];
        v2f bA; bA.x = sW1[kb][lm];        bA.y = sW1[kb + 1][lm];
        v2f bB; bB.x = sW1[kb][16 + lm];   bB.y = sW1[kb + 1][16 + lm];
        acc0 = __builtin_amdgcn_wmma_f32_16x16x4_f32(false, a, false, bA, (short)0, acc0, false, false);
        acc1 = __builtin_amdgcn_wmma_f32_16x16x4_f32(false, a, false, bB, (short)0, acc1, false, false);
    }
    // ReLU + stage h1 row-major for reuse as next A-matrix (same-wave LDS is in-order)
    #pragma unroll
    for (int v = 0; v < 8; ++v) {
        sH[m0 + v + 8 * hi][lm]      = fmaxf(acc0[v], 0.0f);
        sH[m0 + v + 8 * hi][16 + lm] = fmaxf(acc1[v], 0.0f);
    }

    // GEMM2: h1(16x32) @ W2(32x32)
    v8f acc2 = {}, acc3 = {};
    for (int kk = 0; kk < HID / 4; ++kk) {
        int kb = 4 * kk + 2 * hi;
        v2f a;  a.x  = sH[m0 + lm][kb];    a.y  = sH[m0 + lm][kb + 1];
        v2f bA; bA.x = sW2[kb][lm];        bA.y = sW2[kb + 1][lm];
        v2f bB; bB.x = sW2[kb][16 + lm];   bB.y = sW2[kb + 1][16 + lm];
        acc2 = __builtin_amdgcn_wmma_f32_16x16x4_f32(false, a, false, bA, (short)0, acc2, false, false);
        acc3 = __builtin_amdgcn_wmma_f32_16x16x4_f32(false, a, false, bB, (short)0, acc3, false, false);
    }
    #pragma unroll
    for (int v = 0; v < 8; ++v) {
        sH[m0 + v + 8 * hi][lm]      = fmaxf(acc2[v], 0.0f);
        sH[m0 + v + 8 * hi][16 + lm] = fmaxf(acc3[v], 0.0f);
    }

    // GEMM3: h2(16x32) @ W3(32x1) via split-lane dot + cross-half shuffle
    {
        int row = m0 + lm;
        float part = 0.0f;
        #pragma unroll
        for (int k = 0; k < 16; ++k)
            part += sH[row][hi * 16 + k] * sW3[hi * 16 + k];
        part += __shfl_xor(part, 16, 32);
        if (hi == 0) {
            int b = b0 + row;
            float nr = part + b3[0];
            float ir = sIR[row];
            out[b]         = ir + nr;   // symbolic_val
            out[(size_t)B + b]     = ir; // immediate_reward
            out[(size_t)2 * B + b] = nr; // next_return
        }
    }
}

extern "C" void kernel_launch(void* const* d_in, const int* in_sizes, int n_in,
                              void* d_out, int out_size, void* d_ws, size_t ws_size,
                              hipStream_t stream) {
    const float* obs = (const float*)d_in[0];
    const float* ac  = (const float*)d_in[1];
    const float* W1  = (const float*)d_in[2];
    const float* W2  = (const float*)d_in[3];
    const float* W3  = (const float*)d_in[4];
    const float* b3  = (const float*)d_in[5];
    float* out = (float*)d_out;

    int B = in_sizes[0] / (2 * SN);          // 16384
    int blocks = (B + ROWS_PER_BLOCK - 1) / ROWS_PER_BLOCK;
    grid_vpn_fused<<<blocks, THREADS, 0, stream>>>(obs, ac, W1, W2, W3, b3, out, B);
}